// GCN_IB_81217831567575
// MI455X (gfx1250) — compile-verified
//
#include <hip/hip_runtime.h>
#include <hip/hip_bf16.h>
#include <math.h>

// ---------------- problem constants ----------------
#define NN   8192
#define EE   262144
#define FIN  512
#define HH   16
#define CC   7

// d_out float offsets (return-order concat)
#define O_ADJ   0
#define O_NF2   67108864          // N*N
#define O_GE    67166208          // +N*7
#define O_POS   67166215
#define O_NEG   67166222
#define O_PP    67166229
#define O_LOGP  67166230

typedef __attribute__((ext_vector_type(2))) float v2f;
typedef __attribute__((ext_vector_type(8))) float v8f;

// ---------------- kernels ----------------

// deg init (=1.0 for self loop) + zero small accumulators (group[14] + nag[4])
__global__ void k_init(float* __restrict__ deg, float* __restrict__ grp) {
    int i = blockIdx.x * blockDim.x + threadIdx.x;
    if (i < NN) deg[i] = 1.0f;
    if (i < 18) grp[i] = 0.0f;      // grp[0..13]=group, grp[14..17]=new_adj_global
}

// deg[col] += w
__global__ void k_deg(const int* __restrict__ ei, const float* __restrict__ w,
                      float* __restrict__ deg) {
    int e = blockIdx.x * blockDim.x + threadIdx.x;
    int c = ei[EE + e];
    atomicAdd(&deg[c], w[e]);
}

// dinv = deg^-1/2 (in place)
__global__ void k_dinv(float* __restrict__ d) {
    int i = blockIdx.x * blockDim.x + threadIdx.x;
    float v = d[i];
    d[i] = (v > 0.0f) ? rsqrtf(v) : 0.0f;
}

// hw1 = x @ W1  via V_WMMA_F32_16X16X4_F32.  One wave per 16x16 tile, K=512.
__global__ void k_gemm1(const float* __restrict__ x, const float* __restrict__ W1,
                        float* __restrict__ hw1) {
    int wave = blockIdx.x * (blockDim.x >> 5) + (threadIdx.x >> 5);
    int lane = threadIdx.x & 31;
    int rowBase = wave * 16;                 // 512 waves total, exact
    int m    = lane & 15;                    // row (A) / col (B)
    int koff = (lane >> 4) * 2;              // 0 or 2
    const float* xr = x + (size_t)(rowBase + m) * FIN + koff;
    v8f acc = {};
    for (int k0 = 0; k0 < FIN; k0 += 4) {
        v2f a, b;
        a.x = xr[k0];                        // K = k0+koff
        a.y = xr[k0 + 1];                    // K = k0+koff+1
        b.x = W1[(k0 + koff) * HH + m];      // B row K, col n
        b.y = W1[(k0 + koff + 1) * HH + m];
        acc = __builtin_amdgcn_wmma_f32_16x16x4_f32(
            false, a, false, b, (short)0, acc, false, false);
    }
    int mHi = (lane >> 4) * 8;
    for (int r = 0; r < 8; ++r)
        hw1[(size_t)(rowBase + mHi + r) * HH + m] = acc[r];
}

// nf1 = self-loop contribution: hw1[i]*dinv[i]^2   (also serves as init)
__global__ void k_conv1_init(const float* __restrict__ hw1, const float* __restrict__ dinv,
                             float* __restrict__ nf1) {
    int idx = blockIdx.x * blockDim.x + threadIdx.x;   // N*16
    int i = idx >> 4;
    float d = dinv[i];
    nf1[idx] = hw1[idx] * d * d;
}

// edge scatter for conv1: nf1[col] += hw1[row] * dinv[row]*w*dinv[col]
__global__ void k_conv1_scatter(const int* __restrict__ ei, const float* __restrict__ w,
                                const float* __restrict__ dinv, const float* __restrict__ hw1,
                                float* __restrict__ nf1) {
    int idx = blockIdx.x * blockDim.x + threadIdx.x;   // E*16
    int e = idx >> 4, k = idx & 15;
    int r = ei[e], c = ei[EE + e];
    float nrm = dinv[r] * w[e] * dinv[c];
    atomicAdd(&nf1[c * HH + k], hw1[r * HH + k] * nrm);
}

// per node: v = relu(nf1 + b1); hw2 = v @ W2
__global__ void k_node2(const float* __restrict__ nf1, const float* __restrict__ b1,
                        const float* __restrict__ W2, const float* __restrict__ b2_unused,
                        float* __restrict__ hw2) {
    int i = blockIdx.x * blockDim.x + threadIdx.x;     // N
    float v[HH];
    for (int k = 0; k < HH; ++k)
        v[k] = fmaxf(nf1[i * HH + k] + b1[k], 0.0f);
    for (int j = 0; j < CC; ++j) {
        float s = 0.0f;
        for (int k = 0; k < HH; ++k) s += v[k] * W2[k * CC + j];
        hw2[i * CC + j] = s;
    }
}

// nf2(out) = self-loop contribution of conv2
__global__ void k_conv2_init(const float* __restrict__ hw2, const float* __restrict__ dinv,
                             float* __restrict__ nf2) {
    int idx = blockIdx.x * blockDim.x + threadIdx.x;   // N*7
    int i = idx / CC;
    float d = dinv[i];
    nf2[idx] = hw2[idx] * d * d;
}

// edge scatter for conv2 (7 atomics per edge)
__global__ void k_conv2_scatter(const int* __restrict__ ei, const float* __restrict__ w,
                                const float* __restrict__ dinv, const float* __restrict__ hw2,
                                float* __restrict__ nf2) {
    int e = blockIdx.x * blockDim.x + threadIdx.x;     // E
    int r = ei[e], c = ei[EE + e];
    float nrm = dinv[r] * w[e] * dinv[c];
    for (int j = 0; j < CC; ++j)
        atomicAdd(&nf2[c * CC + j], hw2[r * CC + j] * nrm);
}

// nf2 += b2
__global__ void k_conv2_bias(float* __restrict__ nf2, const float* __restrict__ b2) {
    int idx = blockIdx.x * blockDim.x + threadIdx.x;   // N*7
    nf2[idx] += b2[idx % CC];
}

// per node: assignment, logp; block-reduced group accumulation
__global__ void k_assign(const float* __restrict__ nf2,
                         const float* __restrict__ Wf1, const float* __restrict__ bf1,
                         const float* __restrict__ Wf2, const float* __restrict__ bf2,
                         float* __restrict__ asn, float* __restrict__ grp,
                         float* __restrict__ logp) {
    __shared__ float sg[2 * CC];
    int tid = threadIdx.x;
    if (tid < 2 * CC) sg[tid] = 0.0f;
    __syncthreads();

    int i = blockIdx.x * blockDim.x + tid;             // N, exact
    float nf[CC];
    for (int j = 0; j < CC; ++j) nf[j] = nf2[i * CC + j];

    float a1[HH];
    for (int h = 0; h < HH; ++h) {
        float s = bf1[h];
        for (int c = 0; c < CC; ++c) s += nf[c] * Wf1[c * HH + h];
        a1[h] = tanhf(s);
    }
    float z0 = bf2[0], z1 = bf2[1];
    for (int h = 0; h < HH; ++h) { z0 += a1[h] * Wf2[h * 2]; z1 += a1[h] * Wf2[h * 2 + 1]; }
    float mz = fmaxf(z0, z1);
    float e0 = expf(z0 - mz), e1 = expf(z1 - mz);
    float inv = 1.0f / (e0 + e1);
    float p0 = e0 * inv, p1 = e1 * inv;
    asn[i * 2] = p0; asn[i * 2 + 1] = p1;

    // log_softmax over 7 classes (x2 == nf2, see analysis)
    float mm = nf[0];
    for (int j = 1; j < CC; ++j) mm = fmaxf(mm, nf[j]);
    float ss = 0.0f;
    for (int j = 0; j < CC; ++j) ss += expf(nf[j] - mm);
    float ls = logf(ss);
    for (int j = 0; j < CC; ++j) logp[i * CC + j] = nf[j] - mm - ls;

    // group[g][c] += p_g * nf[c]  (LDS reduce, one global atomic per slot per block)
    for (int c = 0; c < CC; ++c) {
        atomicAdd(&sg[c], p0 * nf[c]);
        atomicAdd(&sg[CC + c], p1 * nf[c]);
    }
    __syncthreads();
    if (tid < 2 * CC) atomicAdd(&grp[tid], sg[tid]);
}

// edge scatter into dense new_adj (only rows with keep=1) + new_adj_global reduce
__global__ void k_adj(const int* __restrict__ ei, const float* __restrict__ asn,
                      float* __restrict__ adj, float* __restrict__ nag) {
    __shared__ float sn[4];
    int tid = threadIdx.x;
    if (tid < 4) sn[tid] = 0.0f;
    __syncthreads();

    int e = blockIdx.x * blockDim.x + tid;             // E, exact
    int r = ei[e], c = ei[EE + e];
    float p0r = asn[2 * r], p1r = asn[2 * r + 1];
    float p0c = asn[2 * c], p1c = asn[2 * c + 1];
    atomicAdd(&sn[0], p0r * p0c);
    atomicAdd(&sn[1], p0r * p1c);
    atomicAdd(&sn[2], p1r * p0c);
    atomicAdd(&sn[3], p1r * p1c);
    if (p0r >= p1r)
        atomicAdd(&adj[(size_t)r * NN + c], 1.0f);
    __syncthreads();
    if (tid < 4) atomicAdd(&nag[tid], sn[tid]);
}

// scalars: graph_embedding, positive, negative, pos_penalty
__global__ void k_final(const float* __restrict__ grp, const float* __restrict__ nag,
                        float* __restrict__ out) {
    int t = threadIdx.x;
    if (t < CC) {
        float g0 = grp[t], g1 = grp[CC + t];
        out[O_GE + t]  = 0.5f * (g0 + g1);
        out[O_POS + t] = fminf(fmaxf(g0, -100.0f), 100.0f);
        out[O_NEG + t] = fminf(fmaxf(g1, -100.0f), 100.0f);
    }
    if (t == 0) {
        float a = nag[0], b = nag[1], c = nag[2], d = nag[3];
        float n00 = a / fmaxf(fabsf(a) + fabsf(b), 1e-12f);
        float n11 = d / fmaxf(fabsf(c) + fabsf(d), 1e-12f);
        float d0 = n00 - 1.0f, d1 = n11 - 1.0f;
        out[O_PP] = 0.5f * (d0 * d0 + d1 * d1);
    }
}

// ---------------- launch ----------------
extern "C" void kernel_launch(void* const* d_in, const int* in_sizes, int n_in,
                              void* d_out, int out_size, void* d_ws, size_t ws_size,
                              hipStream_t stream) {
    const float* x   = (const float*)d_in[0];
    const int*   ei  = (const int*)d_in[1];       // [2,E]: rows then cols
    const float* w   = (const float*)d_in[2];
    const float* W1  = (const float*)d_in[3];
    const float* b1  = (const float*)d_in[4];
    const float* W2  = (const float*)d_in[5];
    const float* b2  = (const float*)d_in[6];
    const float* Wf1 = (const float*)d_in[7];
    const float* bf1 = (const float*)d_in[8];
    const float* Wf2 = (const float*)d_in[9];
    const float* bf2 = (const float*)d_in[10];

    float* out = (float*)d_out;
    float* ws  = (float*)d_ws;
    float* dinv = ws;                  // N        (deg, then dinv in place)
    float* hw1  = dinv + NN;           // N*16
    float* nf1  = hw1 + NN * HH;       // N*16
    float* hw2  = nf1 + NN * HH;       // N*7
    float* asn  = hw2 + NN * CC;       // N*2
    float* grp  = asn + NN * 2;        // 14 (group) + 4 (new_adj_global)
    float* nag  = grp + 2 * CC;

    float* nf2  = out + O_NF2;
    float* logp = out + O_LOGP;

    // big dense output: zero then scatter
    hipMemsetAsync(out, 0, (size_t)NN * NN * sizeof(float), stream);

    k_init<<<NN / 256, 256, 0, stream>>>(dinv, grp);
    k_deg<<<EE / 256, 256, 0, stream>>>(ei, w, dinv);
    k_dinv<<<NN / 256, 256, 0, stream>>>(dinv);

    k_gemm1<<<(NN / 16) / 8, 256, 0, stream>>>(x, W1, hw1);   // 8 waves/block

    k_conv1_init<<<NN * HH / 256, 256, 0, stream>>>(hw1, dinv, nf1);
    k_conv1_scatter<<<EE * HH / 256, 256, 0, stream>>>(ei, w, dinv, hw1, nf1);
    k_node2<<<NN / 256, 256, 0, stream>>>(nf1, b1, W2, b2, hw2);

    k_conv2_init<<<NN * CC / 256, 256, 0, stream>>>(hw2, dinv, nf2);
    k_conv2_scatter<<<EE / 256, 256, 0, stream>>>(ei, w, dinv, hw2, nf2);
    k_conv2_bias<<<NN * CC / 256, 256, 0, stream>>>(nf2, b2);

    k_assign<<<NN / 256, 256, 0, stream>>>(nf2, Wf1, bf1, Wf2, bf2, asn, grp, logp);
    k_adj<<<EE / 256, 256, 0, stream>>>(ei, asn, out, nag);
    k_final<<<1, 32, 0, stream>>>(grp, nag, out);
}